// GARNOLDI_89326729822515
// MI455X (gfx1250) — compile-verified
//
#include <hip/hip_runtime.h>
#include <stdint.h>

typedef float v2f __attribute__((ext_vector_type(2)));
typedef float v8f __attribute__((ext_vector_type(8)));

#define D_FEAT 64

#if defined(__has_builtin)
#if __has_builtin(__builtin_amdgcn_wmma_f32_16x16x4_f32)
#define USE_WMMA_AXPY 1
#endif
#endif

// ---------------- degree / CSR construction ----------------

__global__ void deg_init_kernel(float* __restrict__ deg, int N) {
  int i = blockIdx.x * blockDim.x + threadIdx.x;
  if (i < N) deg[i] = 1.0f;  // self loop
}

__global__ void deg_count_kernel(const int* __restrict__ dst, float* __restrict__ deg, int E) {
  int e = blockIdx.x * blockDim.x + threadIdx.x;
  if (e < E) atomicAdd(&deg[dst[e]], 1.0f);
}

// single-block exclusive scan of per-node edge counts -> row_ptr / fill_ptr; also dinv
__global__ __launch_bounds__(1024) void scan_kernel(const float* __restrict__ deg,
                                                    int* __restrict__ row_ptr,
                                                    int* __restrict__ fill_ptr,
                                                    float* __restrict__ dinv, int N) {
  __shared__ int psum[1024];
  int t = threadIdx.x;
  int npt = N >> 10;            // nodes per thread (64 for N=65536)
  int base = t * npt;
  int s = 0;
  for (int j = 0; j < npt; ++j) s += (int)deg[base + j] - 1;
  psum[t] = s;
  __syncthreads();
  for (int off = 1; off < 1024; off <<= 1) {
    int v = (t >= off) ? psum[t - off] : 0;
    __syncthreads();
    psum[t] += v;
    __syncthreads();
  }
  int run = psum[t] - s;        // exclusive prefix for this chunk
  for (int j = 0; j < npt; ++j) {
    int i = base + j;
    row_ptr[i] = run;
    fill_ptr[i] = run;
    run += (int)deg[i] - 1;
    dinv[i] = rsqrtf(deg[i]);
  }
  if (t == 1023) row_ptr[N] = run;
}

__global__ void csr_fill_kernel(const int* __restrict__ src, const int* __restrict__ dst,
                                const float* __restrict__ dinv, int* __restrict__ fill_ptr,
                                int* __restrict__ csr_src, float* __restrict__ csr_w, int E) {
  int e = blockIdx.x * blockDim.x + threadIdx.x;
  if (e >= E) return;
  int s = src[e], d = dst[e];
  int p = atomicAdd(&fill_ptr[d], 1);
  csr_src[p] = s;
  csr_w[p] = dinv[s] * dinv[d];
}

// ---------------- per-hop gather: h_next = A_hat * h_cur ----------------
// One wave per dst node; 32 lanes x float2 = 64 features.
// Neighbor rows are streamed through LDS with async global->LDS copies in a
// 4-deep per-wave software pipeline (1 KB in flight per wave), synchronized
// with s_wait_asynccnt (in-order async completion => oldest slot is ready).

__global__ __launch_bounds__(256) void hop_kernel(const float* __restrict__ h_cur,
                                                  float* __restrict__ h_next,
                                                  const int* __restrict__ row_ptr,
                                                  const int* __restrict__ csr_src,
                                                  const float* __restrict__ csr_w,
                                                  const float* __restrict__ dinv) {
  __shared__ float2 lbuf[8 * 4 * 32];  // 8 waves x 4 slots x 32 lanes (8 KB)
  int lane = threadIdx.x & 31;
  int wib = threadIdx.x >> 5;
  int node = blockIdx.x * 8 + wib;
  const float2* hc = (const float2*)h_cur;

  float di = dinv[node];               // wave-uniform
  float2 self = hc[(size_t)node * 32 + lane];
  float sw = di * di;                  // self-loop norm = 1/deg
  float ax = sw * self.x, ay = sw * self.y;

  int e0 = row_ptr[node], e1 = row_ptr[node + 1];
  // byte address of this lane's element in slot 0 of this wave; slots stride 256 B
  unsigned ldsbase = (unsigned)(uintptr_t)&lbuf[wib * 128 + lane];

  if (e1 > e0) {
    int elast = e1 - 1;
#pragma unroll
    for (int p = 0; p < 3; ++p) {      // preload pipeline (clamped; dups harmless)
      int ep = e0 + p;
      ep = (ep > elast) ? elast : ep;
      unsigned ldsp = ldsbase + (unsigned)p * 256u;
      const float2* gpp = hc + (size_t)csr_src[ep] * 32 + lane;
      asm volatile("global_load_async_to_lds_b64 %0, %1, off" :: "v"(ldsp), "v"(gpp) : "memory");
    }
    for (int e = e0; e < e1; ++e) {
      int en = e + 3;
      en = (en > elast) ? elast : en;
      unsigned ldsn = ldsbase + (unsigned)(((e + 3) - e0) & 3) * 256u;
      const float2* gpn = hc + (size_t)csr_src[en] * 32 + lane;
      asm volatile("global_load_async_to_lds_b64 %0, %1, off" :: "v"(ldsn), "v"(gpn) : "memory");
      asm volatile("s_wait_asynccnt 0x3" ::: "memory");  // edge e's row landed in LDS
      float2 g = lbuf[wib * 128 + ((e - e0) & 3) * 32 + lane];
      float w = csr_w[e];                                // wave-uniform
      ax += w * g.x;
      ay += w * g.y;
    }
    asm volatile("s_wait_asynccnt 0x0" ::: "memory");    // drain before wave exit
  }
  float2 acc;
  acc.x = ax;
  acc.y = ay;
  ((float2*)h_next)[(size_t)node * 32 + lane] = acc;
}

// ---------------- axpy via WMMA: out = a*base + t*h ----------------
// One wave per 16x16 (node x feature) tile. D = C + sum_j A_j x B_j with
// A_j = h[:, 4j:4j+4] (16x4 f32 layout) and B_j = t * I16[4j:4j+4, :].

__global__ __launch_bounds__(256) void axpy_kernel(const float* __restrict__ h,
                                                   const float* basep,
                                                   const float* __restrict__ temp,
                                                   int a_idx, int t_idx,
                                                   float* out) {
  int lane = threadIdx.x & 31;
  int wid = (blockIdx.x * blockDim.x + threadIdx.x) >> 5;
  int nt = wid >> 2, ft = wid & 3;       // 4 feature tiles of 16 in D_FEAT=64
  int node0 = nt * 16, f0 = ft * 16;
  int half = lane >> 4, l = lane & 15;

  float a = (a_idx >= 0) ? temp[a_idx] : 1.0f;
  float t = temp[t_idx];

  v8f c;
#pragma unroll
  for (int r = 0; r < 8; ++r)  // C layout: VGPR r -> (M = r + 8*half, N = l)
    c[r] = a * basep[(size_t)(node0 + r + 8 * half) * D_FEAT + f0 + l];

#ifdef USE_WMMA_AXPY
  int k0 = 2 * half;  // A layout: lanes 0-15 hold K={0,1}, lanes 16-31 K={2,3}; M = l
#pragma unroll
  for (int j = 0; j < 4; ++j) {
    v2f av, bv;
    const float* ap = &h[(size_t)(node0 + l) * D_FEAT + f0 + 4 * j + k0];
    av.x = ap[0];
    av.y = ap[1];
    // B layout mirror: VGPR0 -> row K=k0 (N=l), VGPR1 -> row K=k0+1
    bv.x = (l == 4 * j + k0) ? t : 0.0f;
    bv.y = (l == 4 * j + k0 + 1) ? t : 0.0f;
    c = __builtin_amdgcn_wmma_f32_16x16x4_f32(false, av, false, bv, (short)0, c,
                                              false, false);
  }
#else
#pragma unroll
  for (int r = 0; r < 8; ++r)
    c[r] += t * h[(size_t)(node0 + r + 8 * half) * D_FEAT + f0 + l];
#endif

#pragma unroll
  for (int r = 0; r < 8; ++r)
    out[(size_t)(node0 + r + 8 * half) * D_FEAT + f0 + l] = c[r];
}

// ---------------- host-side orchestration ----------------

extern "C" void kernel_launch(void* const* d_in, const int* in_sizes, int n_in,
                              void* d_out, int out_size, void* d_ws, size_t ws_size,
                              hipStream_t stream) {
  const float* x = (const float*)d_in[0];
  const float* temp = (const float*)d_in[1];
  const int* ei = (const int*)d_in[2];
  const int N = in_sizes[0] / D_FEAT;
  const int E = in_sizes[2] / 2;
  const int K = in_sizes[1] - 1;        // temp has K+1 entries
  const int* src = ei;                  // edge_index[0]
  const int* dst = ei + E;              // edge_index[1]
  float* out = (float*)d_out;
  (void)n_in; (void)out_size; (void)ws_size;

  char* ws = (char*)d_ws;
  size_t off = 0;
  auto alloc = [&](size_t bytes) -> char* {
    char* p = ws + off;
    off += (bytes + 255) & ~(size_t)255;
    return p;
  };
  float* bufA = (float*)alloc((size_t)N * D_FEAT * sizeof(float));
  float* bufB = (float*)alloc((size_t)N * D_FEAT * sizeof(float));
  int* csr_src = (int*)alloc((size_t)E * sizeof(int));
  float* csr_w = (float*)alloc((size_t)E * sizeof(float));
  int* row_ptr = (int*)alloc((size_t)(N + 1) * sizeof(int));
  int* fill_ptr = (int*)alloc((size_t)N * sizeof(int));
  float* deg = (float*)alloc((size_t)N * sizeof(float));
  float* dinv = (float*)alloc((size_t)N * sizeof(float));

  // Build GCN-normalized CSR (by dst) once per call.
  deg_init_kernel<<<(N + 255) / 256, 256, 0, stream>>>(deg, N);
  deg_count_kernel<<<(E + 255) / 256, 256, 0, stream>>>(dst, deg, E);
  scan_kernel<<<1, 1024, 0, stream>>>(deg, row_ptr, fill_ptr, dinv, N);
  csr_fill_kernel<<<(E + 255) / 256, 256, 0, stream>>>(src, dst, dinv, fill_ptr,
                                                       csr_src, csr_w, E);

  // K-1 propagation hops; hop m uses coefficient temp[K-2-m].
  // First axpy also applies hidden init temp[K-1]*x.
  for (int m = 0; m < K - 1; ++m) {
    int k = K - 2 - m;
    const float* hcur = (m == 0) ? x : ((m & 1) ? bufA : bufB);
    float* hnext = (m & 1) ? bufB : bufA;
    hop_kernel<<<N / 8, 256, 0, stream>>>(hcur, hnext, row_ptr, csr_src, csr_w, dinv);
    axpy_kernel<<<(N / 16) * (D_FEAT / 16) / 8, 256, 0, stream>>>(
        hnext, (m == 0) ? x : out, temp, (m == 0) ? (K - 1) : -1, k, out);
  }
}